// PrimaryCaps_71880572665969
// MI455X (gfx1250) — compile-verified
//
#include <hip/hip_runtime.h>
#include <math.h>

typedef __attribute__((ext_vector_type(2))) float v2f;
typedef __attribute__((ext_vector_type(4))) float v4f;
typedef __attribute__((ext_vector_type(8))) float v8f;

#define BATCH   16384
#define IN_CH   768
#define NUM_N   512          // 8 caps * 64 routes
#define TILE_M  128
#define TILE_N  64
#define KC      64
#define LDS_STRIDE 68        // 64 + 4 floats pad -> bank-conflict-free, 272B rows (16B aligned)

// GEMM: out[b, mod*64 + r, c] = sum_k x[b,k] * W[c*64+r, k] + bias[c*64+r]
// N-tile of 64 == one capsule c = blockIdx.y.
__global__ __launch_bounds__(256) void caps_gemm_kernel(
    const float* __restrict__ img,  const float* __restrict__ capt, const float* __restrict__ dct,
    const float* __restrict__ Wimg, const float* __restrict__ Wcapt, const float* __restrict__ Wdct,
    const float* __restrict__ bimg, const float* __restrict__ bcapt, const float* __restrict__ bdct,
    float* __restrict__ out)
{
    __shared__ __align__(16) float Bs[TILE_N * LDS_STRIDE];

    const int mod = blockIdx.z;
    const float* x    = (mod == 0) ? img  : (mod == 1) ? capt  : dct;
    const float* W    = (mod == 0) ? Wimg : (mod == 1) ? Wcapt : Wdct;
    const float* bias = (mod == 0) ? bimg : (mod == 1) ? bcapt : bdct;

    const int tid  = threadIdx.x;
    const int wave = tid >> 5;        // 0..7
    const int lane = tid & 31;
    const int lr   = lane & 15;       // fragment row/col index
    const int lh   = lane >> 4;       // 0/1 -> K pair select

    const int rowBase = blockIdx.x * TILE_M;   // batch rows
    const int cap     = blockIdx.y;            // capsule 0..7
    const int nBase   = cap * TILE_N;          // flat n base

    // 4 N-subtiles of 16 -> 4 x v8f accumulators (32 VGPRs)
    v8f acc[4] = {};

    // per-lane A pointer: row = rowBase + wave*16 + lr, k offset = lh*2
    const float* aPtr  = x + (size_t)(rowBase + wave * 16 + lr) * IN_CH + lh * 2;
    const float* wBase = W + (size_t)nBase * IN_CH;

    for (int k0 = 0; k0 < IN_CH; k0 += KC) {
        // Cooperative B-tile fill: 64 rows x 64 k-cols of W, float4 per thread x4
        #pragma unroll
        for (int j = 0; j < 4; ++j) {
            int idx  = j * 256 + tid;        // 0..1023
            int row  = idx >> 4;             // 0..63
            int colv = (idx & 15) << 2;      // 0,4,...,60
            v4f val = *(const v4f*)(wBase + (size_t)row * IN_CH + k0 + colv);
            *(v4f*)(&Bs[row * LDS_STRIDE + colv]) = val;
        }
        __syncthreads();

        if (k0 + KC < IN_CH)
            __builtin_prefetch(aPtr + k0 + KC, 0, 1);   // global_prefetch_b8

        #pragma unroll
        for (int kk = 0; kk < KC; kk += 4) {
            // A fragment 16x4 fp32: lane lr row, K = kk + lh*2 + {0,1}
            v2f a = *(const v2f*)(aPtr + k0 + kk);
            #pragma unroll
            for (int t = 0; t < 4; ++t) {
                // B fragment 4x16: lane lr = column n, K = kk + lh*2 + {0,1}
                v2f b = *(const v2f*)(&Bs[(t * 16 + lr) * LDS_STRIDE + kk + lh * 2]);
                acc[t] = __builtin_amdgcn_wmma_f32_16x16x4_f32(
                    /*neg_a=*/false, a, /*neg_b=*/false, b,
                    /*c_mod=*/(short)0, acc[t], /*reuse_a=*/false, /*reuse_b=*/false);
            }
        }
        __syncthreads();
    }

    // Epilogue: bias + scattered store into [B, 192, 8]
    #pragma unroll
    for (int t = 0; t < 4; ++t) {
        const float bv = bias[nBase + t * 16 + lr];     // depends only on column n
        const int r  = t * 16 + lr;                     // route 0..63
        const int rr = mod * 64 + r;                    // 0..191
        #pragma unroll
        for (int v = 0; v < 8; ++v) {
            const int brow = rowBase + wave * 16 + v + lh * 8;   // C/D layout: M = v (+8 for hi lanes)
            out[(size_t)brow * (192 * 8) + (size_t)rr * 8 + cap] = acc[t][v] + bv;
        }
    }
}

// In-place squash over the 8-capsule vector (contiguous innermost dim)
__global__ __launch_bounds__(256) void squash_kernel(float* __restrict__ out, int ngroups)
{
    int g = blockIdx.x * blockDim.x + threadIdx.x;
    if (g >= ngroups) return;
    v4f* p = (v4f*)(out + (size_t)g * 8);
    v4f u0 = p[0];
    v4f u1 = p[1];
    float sq = u0.x*u0.x + u0.y*u0.y + u0.z*u0.z + u0.w*u0.w
             + u1.x*u1.x + u1.y*u1.y + u1.z*u1.z + u1.w*u1.w;
    float scale = sq / ((1.0f + sq) * sqrtf(sq + 1e-7f));
    p[0] = u0 * scale;
    p[1] = u1 * scale;
}

extern "C" void kernel_launch(void* const* d_in, const int* in_sizes, int n_in,
                              void* d_out, int out_size, void* d_ws, size_t ws_size,
                              hipStream_t stream)
{
    const float* img   = (const float*)d_in[0];
    const float* capt  = (const float*)d_in[1];
    const float* dct   = (const float*)d_in[2];
    const float* Wimg  = (const float*)d_in[3];
    const float* bimg  = (const float*)d_in[4];
    const float* Wcapt = (const float*)d_in[5];
    const float* bcapt = (const float*)d_in[6];
    const float* Wdct  = (const float*)d_in[7];
    const float* bdct  = (const float*)d_in[8];
    float* out = (float*)d_out;

    dim3 grid(BATCH / TILE_M, NUM_N / TILE_N, 3);   // 128 x 8 x 3
    caps_gemm_kernel<<<grid, 256, 0, stream>>>(img, capt, dct,
                                               Wimg, Wcapt, Wdct,
                                               bimg, bcapt, bdct, out);

    const int ngroups = BATCH * 192;                // 3,145,728 capsule vectors
    squash_kernel<<<(ngroups + 255) / 256, 256, 0, stream>>>(out, ngroups);
}